// RRNNforGRUCell_3693671874624
// MI455X (gfx1250) — compile-verified
//
#include <hip/hip_runtime.h>
#include <math.h>

// ---------------------------------------------------------------------------
// RRNN-for-GRU cell on MI455X (gfx1250). Bandwidth-bound GEMV chain.
//   - 11 batched matvecs v[1,H] x M[4,H,H], H=4096, f32 (exact precision).
//   - Roofline: ~2.9 GB mandatory weight traffic / 23.3 TB/s ≈ 125 us floor;
//     FLOPs (~1.5 G) are free, so WMMA serves as the accumulation engine.
//   - GEMV inner loop uses V_WMMA_F32_16X16X4_F32 (wave32, 16 cols/wave).
//   - Broadcast vector chunk staged to LDS with the Tensor Data Mover.
//   - Deterministic chunked reduction (no float atomics).
// ---------------------------------------------------------------------------

#define H    4096
#define NCH  8                 // h-dimension chunks per GEMV
#define RPC  (H / NCH)         // 512 rows per chunk
#define CPB  128               // columns per block (8 waves x 16 cols)
#define NT   (H / CPB)         // 32 column tiles

typedef __attribute__((ext_vector_type(2))) float        v2f;
typedef __attribute__((ext_vector_type(8))) float        v8f;
typedef __attribute__((ext_vector_type(4))) unsigned int u32x4;
typedef __attribute__((ext_vector_type(8))) int          i32x8;
typedef __attribute__((ext_vector_type(4))) int          i32x4;

#ifndef ENABLE_TDM
#define ENABLE_TDM 1
#endif

// ---------------------------------------------------------------------------
// out_part[ch][l][k] = sum_{h in chunk ch} v[h] * M[l][h][k]
// One wave computes 16 columns via WMMA f32 16x16x4:
//   A[m,k] = v[h0+k] replicated over m (all D rows identical == y tile)
//   B[k,n] = M[h0+k, n0+n]
// A/B f32 16x4 / 4x16 striping (ISA 7.12.2):
//   VGPR0: lanes 0-15 -> K=0, lanes 16-31 -> K=2 ; VGPR1: K=1 / K=3.
// ---------------------------------------------------------------------------
__global__ __launch_bounds__(256) void k_gemv(const float* __restrict__ v,
                                              const float* __restrict__ M,
                                              float* __restrict__ part) {
  __shared__ float lv[RPC];                       // 2 KB vector slice
  const int l    = blockIdx.y;
  const int ch   = blockIdx.z;
  const int t    = threadIdx.x;
  const int wave = t >> 5;
  const int lane = t & 31;
  const int rowbase = ch * RPC;

#if ENABLE_TDM && defined(__has_builtin) && __has_builtin(__builtin_amdgcn_tensor_load_to_lds)
  {
    // Tensor DMA: 1-row tile of RPC f32 elements, global -> LDS.
    unsigned long long ga = (unsigned long long)(const void*)(v + rowbase);
    unsigned int ldsoff   = (unsigned int)(size_t)(void*)lv;
    u32x4 g0;
    g0[0] = 1u;                                            // count=1, user mode
    g0[1] = ldsoff;                                        // lds_addr (bytes)
    g0[2] = (unsigned int)(ga & 0xFFFFFFFFu);              // global_addr[31:0]
    g0[3] = (unsigned int)((ga >> 32) & 0x01FFFFFFu)       // global_addr[56:32]
          | (2u << 30);                                    // type=2 ("image")
    i32x8 g1;
    g1[0] = 0x00020000;                                    // data_size=4B, mask=0
    g1[1] = (int)((unsigned)RPC << 16);                    // tensor_dim0[15:0]
    g1[2] = (int)(1u << 16);                               // tensor_dim1=1
    g1[3] = (int)((unsigned)RPC << 16);                    // tile_dim0=RPC
    g1[4] = 1;                                             // tile_dim1=1
    g1[5] = RPC;                                           // tensor_dim0_stride
    g1[6] = 0;
    g1[7] = 0;
    i32x4 gz;
    gz[0] = 0; gz[1] = 0; gz[2] = 0; gz[3] = 0;
    i32x8 gz8;
    gz8[0] = 0; gz8[1] = 0; gz8[2] = 0; gz8[3] = 0;
    gz8[4] = 0; gz8[5] = 0; gz8[6] = 0; gz8[7] = 0;
    // amdgpu-toolchain (clang-23) 6-arg form:
    // (g0, g1, g2, g3, extra_group, cpol)
    __builtin_amdgcn_tensor_load_to_lds(g0, g1, gz, gz, gz8, 0);
    __builtin_amdgcn_s_wait_tensorcnt(0);                  // per-wave completion
  }
#else
  for (int i = t; i < RPC; i += 256) lv[i] = v[rowbase + i];
#endif
  __syncthreads();

  const int n0   = blockIdx.x * CPB + wave * 16;
  const int col  = n0 + (lane & 15);
  const int koff = ((lane >> 4) & 1) * 2;                  // 0 for lanes 0-15, 2 for 16-31
  const float* __restrict__ Mp =
      M + (size_t)l * H * H + ((size_t)rowbase + (size_t)koff) * H + col;

  v8f acc = {};
  for (int it = 0; it < RPC / 4; ++it) {
    const float* __restrict__ p = Mp + (size_t)it * 4u * H;
    v2f bm, a;
    bm.x = p[0];                                           // row k = koff
    bm.y = p[H];                                           // row k = koff+1
    const int k4 = it * 4 + koff;
    a.x = lv[k4];                                          // broadcast v values
    a.y = lv[k4 + 1];
    acc = __builtin_amdgcn_wmma_f32_16x16x4_f32(
        /*neg_a=*/false, a, /*neg_b=*/false, bm,
        /*c_mod=*/(short)0, acc, /*reuse_a=*/false, /*reuse_b=*/false);
  }
  // D row M=0 lives in VGPR0, lanes 0-15 (N = lane).
  if (lane < 16) part[((size_t)ch * 4 + l) * H + n0 + lane] = acc[0];
}

// dense[l][k] = sum_ch part[ch][l][k]  (deterministic reduction)
__global__ __launch_bounds__(256) void k_reduce(const float* __restrict__ part,
                                                float* __restrict__ dense) {
  const int i = blockIdx.x * 256 + threadIdx.x;            // 0 .. 4*H-1
  float s = 0.f;
  for (int c = 0; c < NCH; ++c) s += part[(size_t)c * 4 * H + i];
  dense[i] = s;
}

// z1 = sigmoid(xL[0]+hR[0]+b[0]); r = sigmoid(xL[1]+hR[1]+b[1])
__global__ __launch_bounds__(256) void k_gates(const float* __restrict__ xL,
                                               const float* __restrict__ hR,
                                               const float* __restrict__ b,
                                               float* __restrict__ z1,
                                               float* __restrict__ r) {
  const int h = blockIdx.x * 256 + threadIdx.x;
  z1[h] = 1.f / (1.f + expf(-(xL[h] + hR[h] + b[h])));
  r[h]  = 1.f / (1.f + expf(-(xL[H + h] + hR[H + h] + b[H + h])));
}

// candidate builder: mode 0: A*B+b  1: tanh(A+B+b)  2: 1-(B+b)  3: A+B+b
__global__ __launch_bounds__(256) void k_cand(int mode,
                                              const float* __restrict__ A,
                                              const float* __restrict__ Bm,
                                              const float* __restrict__ b,
                                              float* __restrict__ cand) {
  const int i = blockIdx.x * 256 + threadIdx.x;            // 0 .. 4*H-1
  const float bb = b[i];
  float o;
  if      (mode == 0) o = A[i] * Bm[i] + bb;
  else if (mode == 1) o = tanhf(A[i] + Bm[i] + bb);
  else if (mode == 2) o = 1.f - (Bm[i] + bb);
  else                o = A[i] + Bm[i] + bb;
  cand[i] = o;
}

// s[l] = <cand[l], W_score>; softmax weights -> wmix; argmax + margin -> out
__global__ __launch_bounds__(256) void k_score(const float* __restrict__ cand,
                                               const float* __restrict__ Wsc,
                                               float* __restrict__ wmix,
                                               float* __restrict__ out,
                                               int step) {
  __shared__ float red[4][256];
  const int t = threadIdx.x;
  float s0 = 0.f, s1 = 0.f, s2 = 0.f, s3 = 0.f;
  for (int h = t; h < H; h += 256) {
    const float wv = Wsc[h];
    s0 += cand[h] * wv;
    s1 += cand[H + h] * wv;
    s2 += cand[2 * H + h] * wv;
    s3 += cand[3 * H + h] * wv;
  }
  red[0][t] = s0; red[1][t] = s1; red[2][t] = s2; red[3][t] = s3;
  __syncthreads();
  for (int off = 128; off > 0; off >>= 1) {
    if (t < off) {
      red[0][t] += red[0][t + off];
      red[1][t] += red[1][t + off];
      red[2][t] += red[2][t + off];
      red[3][t] += red[3][t + off];
    }
    __syncthreads();
  }
  if (t == 0) {
    float s[4] = {red[0][0], red[1][0], red[2][0], red[3][0]};
    int idx = 0; float m1 = s[0];
    for (int l = 1; l < 4; ++l) if (s[l] > m1) { m1 = s[l]; idx = l; }
    float m2 = -INFINITY;
    for (int l = 0; l < 4; ++l) if (l != idx && s[l] > m2) m2 = s[l];
    float e[4], esum = 0.f;
    for (int l = 0; l < 4; ++l) { e[l] = expf(s[l] - m1); esum += e[l]; }
    for (int l = 0; l < 4; ++l) wmix[l] = e[l] / esum;
    out[4099 + step] = (float)idx;                         // G_structure[3+step]
    out[4105 + step] = m1 - m2;                            // margins[step]
  }
}

// o[h] = sum_l wmix[l] * cand[l][h]
__global__ __launch_bounds__(256) void k_mix(const float* __restrict__ cand,
                                             const float* __restrict__ wmix,
                                             float* __restrict__ o) {
  const int h = blockIdx.x * 256 + threadIdx.x;
  o[h] = wmix[0] * cand[h] + wmix[1] * cand[H + h] +
         wmix[2] * cand[2 * H + h] + wmix[3] * cand[3 * H + h];
}

// constant head of G_structure = [0, 1, 0]
__global__ void k_init(float* __restrict__ out) {
  out[4096] = 0.f; out[4097] = 1.f; out[4098] = 0.f;
}

// ---------------------------------------------------------------------------
extern "C" void kernel_launch(void* const* d_in, const int* in_sizes, int n_in,
                              void* d_out, int out_size, void* d_ws, size_t ws_size,
                              hipStream_t stream) {
  (void)in_sizes; (void)n_in; (void)out_size; (void)ws_size;
  const float* x   = (const float*)d_in[0];
  const float* hp  = (const float*)d_in[1];
  const float* L   = (const float*)d_in[2];
  const float* R   = (const float*)d_in[3];
  const float* b   = (const float*)d_in[4];
  const float* Wsc = (const float*)d_in[5];
  float* out = (float*)d_out;

  float* wsf  = (float*)d_ws;
  float* part = wsf;                       // NCH*4*H
  float* xL   = part + (size_t)NCH * 4 * H;
  float* hL   = xL  + 4 * H;
  float* hR   = hL  + 4 * H;
  float* mmA  = hR  + 4 * H;
  float* mmB  = mmA + 4 * H;
  float* cand = mmB + 4 * H;
  float* z1   = cand + 4 * H;
  float* r    = z1  + H;
  float* rh   = r   + H;
  float* ht   = rh  + H;
  float* om   = ht  + H;
  float* zht  = om  + H;
  float* z2h  = zht + H;
  float* wmix = z2h + H;                   // 4 floats

  const dim3 gg(NT, 4, NCH), gb(256);
  const dim3 eg(4 * H / 256), hg(H / 256), eb(256);

  auto GEMV = [&](const float* v, const float* M, float* dense) {
    k_gemv<<<gg, gb, 0, stream>>>(v, M, part);
    k_reduce<<<eg, eb, 0, stream>>>(part, dense);
  };

  // xL = x @ L ; hL = h_prev @ L ; hR = h_prev @ R
  GEMV(x,  L, xL);
  GEMV(hp, L, hL);
  GEMV(hp, R, hR);
  k_gates<<<hg, eb, 0, stream>>>(xL, hR, b, z1, r);        // z1 (== z2), r
  k_init<<<1, 1, 0, stream>>>(out);

  // step 0: rh = mix(hL * (r@R) + b)
  GEMV(r, R, mmB);
  k_cand<<<eg, eb, 0, stream>>>(0, hL, mmB, b, cand);
  k_score<<<1, 256, 0, stream>>>(cand, Wsc, wmix, out, 0);
  k_mix<<<hg, eb, 0, stream>>>(cand, wmix, rh);

  // step 1: h_tilde = mix(tanh(xL + rh@R + b))
  GEMV(rh, R, mmB);
  k_cand<<<eg, eb, 0, stream>>>(1, xL, mmB, b, cand);
  k_score<<<1, 256, 0, stream>>>(cand, Wsc, wmix, out, 1);
  k_mix<<<hg, eb, 0, stream>>>(cand, wmix, ht);

  // step 2: oneMinusZ1 = mix(1 - (z1@R + b))
  GEMV(z1, R, mmB);
  k_cand<<<eg, eb, 0, stream>>>(2, xL /*unused*/, mmB, b, cand);
  k_score<<<1, 256, 0, stream>>>(cand, Wsc, wmix, out, 2);
  k_mix<<<hg, eb, 0, stream>>>(cand, wmix, om);

  // step 3: zh_tilde = mix((h_tilde@L) * (oneMinusZ1@R) + b)
  GEMV(ht, L, mmA);
  GEMV(om, R, mmB);
  k_cand<<<eg, eb, 0, stream>>>(0, mmA, mmB, b, cand);
  k_score<<<1, 256, 0, stream>>>(cand, Wsc, wmix, out, 3);
  k_mix<<<hg, eb, 0, stream>>>(cand, wmix, zht);

  // step 4: z2h = mix(hL * (z2@R) + b)   (z2 == z1)
  GEMV(z1, R, mmB);
  k_cand<<<eg, eb, 0, stream>>>(0, hL, mmB, b, cand);
  k_score<<<1, 256, 0, stream>>>(cand, Wsc, wmix, out, 4);
  k_mix<<<hg, eb, 0, stream>>>(cand, wmix, z2h);

  // step 5: h_next = mix((zh_tilde@L) + (z2h@R) + b) -> d_out[0..H-1]
  GEMV(zht, L, mmA);
  GEMV(z2h, R, mmB);
  k_cand<<<eg, eb, 0, stream>>>(3, mmA, mmB, b, cand);
  k_score<<<1, 256, 0, stream>>>(cand, Wsc, wmix, out, 5);
  k_mix<<<hg, eb, 0, stream>>>(cand, wmix, out);
}